// adaptiveAugmentation_25555055411497
// MI455X (gfx1250) — compile-verified
//
#include <hip/hip_runtime.h>
#include <hip/hip_bf16.h>

#define B_ 8
#define C_ 64
#define M_ 4
#define N_ 400
#define T_ 24
#define Q_ 8
#define PPB (M_*N_*T_)        // 38400 points per batch
#define NM (B_*N_*M_)         // 12800
#define MASK_NUM 1280         // int(8*400*4*0.1)

typedef __attribute__((ext_vector_type(2))) float v2f;
typedef __attribute__((ext_vector_type(8))) float v8f;

// ---------------- kernel 1: attention -> softmax -> sim -> w, keep init ----
__global__ void aa_sim_kernel(const float* __restrict__ rep,
                              const float* __restrict__ att_w,
                              const float* __restrict__ att_b,
                              float* __restrict__ w, float* __restrict__ keep) {
  int tt = blockIdx.x * blockDim.x + threadIdx.x;
  if (tt >= B_ * N_) return;
  int b = tt / N_, n = tt % N_;
  float acc[Q_][M_];
#pragma unroll
  for (int q = 0; q < Q_; q++)
#pragma unroll
    for (int m = 0; m < M_; m++) acc[q][m] = att_b[q];
  const float* rb = rep + (size_t)b * C_ * M_ * N_;   // rep[b,c,m,n]
  for (int c = 0; c < C_; c++) {
    float rv[M_];
#pragma unroll
    for (int m = 0; m < M_; m++) rv[m] = rb[(c * M_ + m) * N_ + n];
#pragma unroll
    for (int q = 0; q < Q_; q++) {
      float wq = att_w[q * C_ + c];
#pragma unroll
      for (int m = 0; m < M_; m++) acc[q][m] = fmaf(wq, rv[m], acc[q][m]);
    }
  }
  float smm[M_] = {0.f, 0.f, 0.f, 0.f};
#pragma unroll
  for (int q = 0; q < Q_; q++) {
    float mx = acc[q][0];
#pragma unroll
    for (int m = 1; m < M_; m++) mx = fmaxf(mx, acc[q][m]);
    float e[M_], s = 0.f;
#pragma unroll
    for (int m = 0; m < M_; m++) { e[m] = expf(acc[q][m] - mx); s += e[m]; }
    float inv = 1.f / s;
#pragma unroll
    for (int m = 0; m < M_; m++) smm[m] += e[m] * inv;
  }
#pragma unroll
  for (int m = 0; m < M_; m++) {
    int i = (b * N_ + n) * M_ + m;                    // sim.reshape(-1) order [b,n,m]
    w[i] = 1.0f - smm[m] * (1.0f / Q_);
    keep[i] = 1.0f;                                   // reset mask every call
  }
}

// ---------------- kernel 2: single-block inclusive scan of w -> cw ---------
__global__ void aa_scan_kernel(const float* __restrict__ w, float* __restrict__ cw) {
  __shared__ float s[256];
  int tid = threadIdx.x;
  float carry = 0.f;
  for (int chunk = 0; chunk < NM / 256; chunk++) {
    int i = chunk * 256 + tid;
    s[tid] = w[i];
    __syncthreads();
    for (int off = 1; off < 256; off <<= 1) {
      float t = (tid >= off) ? s[tid - off] : 0.f;
      __syncthreads();
      s[tid] += t;
      __syncthreads();
    }
    cw[i] = s[tid] + carry;
    carry += s[255];
    __syncthreads();
  }
}

// ---------------- kernel 3: threefry2x32 weighted choice -> keep[idx]=0 ----
__device__ __forceinline__ unsigned rotl32(unsigned x, int d) {
  return (x << d) | (x >> (32 - d));
}
__device__ void threefry2x32(unsigned k0, unsigned k1, unsigned x0, unsigned x1,
                             unsigned* o0, unsigned* o1) {
  unsigned ks0 = k0, ks1 = k1, ks2 = k0 ^ k1 ^ 0x1BD11BDAu;
  const int rA[4] = {13, 15, 26, 6}, rB[4] = {17, 29, 16, 24};
  x0 += ks0; x1 += ks1;
#pragma unroll
  for (int i = 0; i < 4; i++) { x0 += x1; x1 = rotl32(x1, rA[i]); x1 ^= x0; }
  x0 += ks1; x1 += ks2 + 1u;
#pragma unroll
  for (int i = 0; i < 4; i++) { x0 += x1; x1 = rotl32(x1, rB[i]); x1 ^= x0; }
  x0 += ks2; x1 += ks0 + 2u;
#pragma unroll
  for (int i = 0; i < 4; i++) { x0 += x1; x1 = rotl32(x1, rA[i]); x1 ^= x0; }
  x0 += ks0; x1 += ks1 + 3u;
#pragma unroll
  for (int i = 0; i < 4; i++) { x0 += x1; x1 = rotl32(x1, rB[i]); x1 ^= x0; }
  x0 += ks1; x1 += ks2 + 4u;
#pragma unroll
  for (int i = 0; i < 4; i++) { x0 += x1; x1 = rotl32(x1, rA[i]); x1 ^= x0; }
  x0 += ks2; x1 += ks0 + 5u;
  *o0 = x0; *o1 = x1;
}

__global__ void aa_choice_kernel(const float* __restrict__ cw, float* __restrict__ keep) {
  int i = blockIdx.x * blockDim.x + threadIdx.x;
  if (i >= MASK_NUM) return;
  const int half = MASK_NUM / 2;  // JAX splits the iota counter in halves
  unsigned x0 = (i < half) ? (unsigned)i : (unsigned)(i - half);
  unsigned x1 = (i < half) ? (unsigned)(i + half) : (unsigned)i;
  unsigned o0, o1;
  threefry2x32(0u, 42u, x0, x1, &o0, &o1);
  unsigned bits = (i < half) ? o0 : o1;
  union { unsigned u; float f; } cv;
  cv.u = (bits >> 9) | 0x3f800000u;                 // [1,2)
  float u01 = cv.f - 1.0f;                          // [0,1)
  float total = cw[NM - 1];
  float r = total * (1.0f - u01);
  int lo = 0, hi = NM;                              // searchsorted(side='left')
  while (lo < hi) {
    int mid = (lo + hi) >> 1;
    if (cw[mid] < r) lo = mid + 1; else hi = mid;
  }
  if (lo >= NM) lo = NM - 1;
  keep[lo] = 0.0f;                                  // identical value; races benign
}

// ------- kernel 4: transposed epilogue tables (o innermost for b128 loads) -
//   WsT [n][o]      = (W2 @ semb)^T                      (400 x 64)
//   WtmT[m][t][o]   = (W2 @ temb)[o,t] + (W2@memb)[o,m] + bias[o]  (4 x 24 x 64)
__global__ void aa_tables_kernel(const float* __restrict__ pw, const float* __restrict__ pb,
                                 const float* __restrict__ temb, const float* __restrict__ semb,
                                 const float* __restrict__ memb,
                                 float* __restrict__ WsT, float* __restrict__ WtmT) {
  int e = blockIdx.x * blockDim.x + threadIdx.x;
  if (e < C_ * N_) {
    int o = e & 63, n = e >> 6;                     // o fastest -> coalesced writes
    float s = 0.f;
    for (int c = 0; c < C_; c++) s = fmaf(pw[o * 2 * C_ + C_ + c], semb[c * N_ + n], s);
    WsT[n * 64 + o] = s;
  } else if (e < C_ * N_ + M_ * T_ * C_) {
    int i = e - C_ * N_;
    int o = i & 63, mt = i >> 6;
    int m2 = mt / T_, t = mt % T_;
    float s = pb[o];
    for (int c = 0; c < C_; c++)
      s = fmaf(pw[o * 2 * C_ + C_ + c], temb[c * T_ + t] + memb[c * M_ + m2], s);
    WtmT[mt * 64 + o] = s;
  }
}

// ---------------- kernel 5: main WMMA GEMM + fused epilogue ----------------
// out[b,o,p] = relu( keep[b,p]*(W1 @ x[b,:,p])[o] + WsT[n,o] + WtmT[m,t,o] )
__global__ __launch_bounds__(256) void aa_main_kernel(
    const float* __restrict__ x, const float* __restrict__ pw,
    const float* __restrict__ keep, const float* __restrict__ WsT,
    const float* __restrict__ WtmT, float* __restrict__ out) {
  __shared__ float sW1[64 * 65];                    // pad stride 65: no LDS bank conflicts
  // stage W1 = proj_w[:, 0:64] (64x64 f32) with b128 loads
  for (int j = threadIdx.x; j < 64 * 16; j += 256) {
    int o = j >> 4, f = j & 15;
    float4 v = *(const float4*)(pw + o * 2 * C_ + f * 4);
    int base = o * 65 + f * 4;
    sW1[base + 0] = v.x; sW1[base + 1] = v.y; sW1[base + 2] = v.z; sW1[base + 3] = v.w;
  }
  __syncthreads();

  int wave = threadIdx.x >> 5;
  int lane = threadIdx.x & 31;
  int gt = blockIdx.x * 8 + wave;                   // 19200 wave-tiles total
  int b = gt / (PPB / 16);
  int tile = gt % (PPB / 16);
  int p0 = tile * 16;
  int pl = p0 + (lane & 15);                        // this lane's column (point)
  int hi = lane >> 4;                               // upper half-wave flag
  int hi2 = hi << 1;                                // K sub-offset 0 / 2

  // B fragments: X[c, p0..p0+15] for all 64 channels (read x exactly once)
  const float* xb = x + (size_t)b * C_ * PPB;
  float bx0[16], bx1[16];
#pragma unroll
  for (int kk = 0; kk < 16; kk++) {
    int c = 4 * kk + hi2;
    bx0[kk] = xb[(size_t)c * PPB + pl];             // VGPR0: K = 4kk + (hi?2:0)
    bx1[kk] = xb[(size_t)(c + 1) * PPB + pl];       // VGPR1: K+1
  }

  int m = pl / (N_ * T_);
  int n = (pl / T_) % N_;
  int t = pl % T_;
  float kp = keep[(b * N_ + n) * M_ + m];           // column scale (applied post-GEMM)
  float* ob = out + (size_t)b * C_ * PPB;
  const float* wsl = WsT + n * 64 + hi * 8;         // this lane's o-run base (32B aligned)
  const float* wtl = WtmT + (m * T_ + t) * 64 + hi * 8;

#pragma unroll
  for (int ot = 0; ot < 4; ot++) {                  // 4 o-tiles of 16 rows, B reused
    int obase = ot * 16;
    int arow = obase + (lane & 15);
    v8f acc = {0.f, 0.f, 0.f, 0.f, 0.f, 0.f, 0.f, 0.f};
#pragma unroll
    for (int kk = 0; kk < 16; kk++) {               // K=64 as 16 x V_WMMA_F32_16X16X4_F32
      int c = 4 * kk + hi2;
      v2f a, bb;
      a.x = sW1[arow * 65 + c];
      a.y = sW1[arow * 65 + c + 1];
      bb.x = bx0[kk];
      bb.y = bx1[kk];
      acc = __builtin_amdgcn_wmma_f32_16x16x4_f32(
          false, a, false, bb, (short)0, acc, false, false);
    }
    // Epilogue: lane's 8 output rows are consecutive o = obase+hi*8+v -> b128 loads
    float4 e0 = *(const float4*)(wsl + obase);
    float4 e1 = *(const float4*)(wsl + obase + 4);
    float4 f0 = *(const float4*)(wtl + obase);
    float4 f1 = *(const float4*)(wtl + obase + 4);
    float ev[8] = {e0.x + f0.x, e0.y + f0.y, e0.z + f0.z, e0.w + f0.w,
                   e1.x + f1.x, e1.y + f1.y, e1.z + f1.z, e1.w + f1.w};
    // C/D layout: VGPR v, lanes0-15 -> row obase+v, lanes16-31 -> row obase+8+v
#pragma unroll
    for (int v = 0; v < 8; v++) {
      int o = obase + hi * 8 + v;
      float val = fmaf(acc[v], kp, ev[v]);
      ob[(size_t)o * PPB + pl] = fmaxf(val, 0.0f);
    }
  }
}

extern "C" void kernel_launch(void* const* d_in, const int* in_sizes, int n_in,
                              void* d_out, int out_size, void* d_ws, size_t ws_size,
                              hipStream_t stream) {
  const float* x      = (const float*)d_in[0];
  const float* rep    = (const float*)d_in[1];
  const float* att_w  = (const float*)d_in[2];
  const float* att_b  = (const float*)d_in[3];
  const float* temb   = (const float*)d_in[4];
  const float* semb   = (const float*)d_in[5];
  const float* memb   = (const float*)d_in[6];
  const float* proj_w = (const float*)d_in[7];
  const float* proj_b = (const float*)d_in[8];

  float* ws   = (float*)d_ws;
  float* w    = ws;                 // 12800
  float* cw   = ws + NM;            // 12800
  float* keep = ws + 2 * NM;        // 12800
  float* WsT  = ws + 3 * NM;        // 25600 (400 x 64)
  float* WtmT = WsT + C_ * N_;      // 6144  (4 x 24 x 64)

  aa_sim_kernel<<<(B_ * N_ + 255) / 256, 256, 0, stream>>>(rep, att_w, att_b, w, keep);
  aa_scan_kernel<<<1, 256, 0, stream>>>(w, cw);
  aa_choice_kernel<<<(MASK_NUM + 255) / 256, 256, 0, stream>>>(cw, keep);
  aa_tables_kernel<<<(C_ * N_ + M_ * T_ * C_ + 255) / 256, 256, 0, stream>>>(
      proj_w, proj_b, temb, semb, memb, WsT, WtmT);
  aa_main_kernel<<<(B_ * PPB / 16) / 8, 256, 0, stream>>>(
      x, proj_w, keep, WsT, WtmT, (float*)d_out);
}